// AudioSegmentHandler_19619410608568
// MI455X (gfx1250) — compile-verified
//
#include <hip/hip_runtime.h>

// Problem constants (fixed by the reference file).
#define B_SZ     32
#define T_LEN    1920000
#define GEN_N    24000
#define N_GAPS   8
#define GAP_LEN  16000
#define CF       1000

// One thread = 4 contiguous output samples (b128 load/store).
// out[b,t] = gen'(t - s_g) for the LAST gap g covering t, else original[b,t].
// gen'(j) = lerp of generated row (24000 -> 16000, align_corners=False) * crossfade(j).
__global__ __launch_bounds__(256) void audio_gap_fill_kernel(
    const float* __restrict__ orig,
    const float* __restrict__ gen,
    const int*   __restrict__ gap_starts,
    float*       __restrict__ out)
{
    const int b   = blockIdx.y;
    const int vec = blockIdx.x * blockDim.x + threadIdx.x;
    const int t0  = vec * 4;
    if (t0 >= T_LEN) return;

    const size_t rowOff = (size_t)b * T_LEN;

    // Streaming copy path: 128-bit load of 4 original samples.
    const float4 v = *reinterpret_cast<const float4*>(orig + rowOff + t0);

    // gfx1250 speculative prefetch 16 KB ahead on the streaming input
    // (global_prefetch_b8; OOB translation failures are silently dropped).
    __builtin_prefetch(orig + rowOff + t0 + 4096, 0, 1);

    // Per-row gap starts: address depends only on blockIdx.y -> scalar loads.
    const int* gs = gap_starts + b * N_GAPS;
    int s[N_GAPS];
#pragma unroll
    for (int g = 0; g < N_GAPS; ++g) s[g] = gs[g];

    // Fast reject: does [t0, t0+4) intersect any gap at all?
    bool any = false;
#pragma unroll
    for (int g = 0; g < N_GAPS; ++g)
        any = any || ((s[g] <= t0 + 3) && (t0 < s[g] + GAP_LEN));

    float4 r = v;
    if (any) {
        const float* __restrict__ grow = gen + (size_t)b * GEN_N;
        float vals[4] = { v.x, v.y, v.z, v.w };
#pragma unroll
        for (int e = 0; e < 4; ++e) {
            const int t = t0 + e;
            int j = -1;
#pragma unroll
            for (int g = 0; g < N_GAPS; ++g) {   // ascending: last hit wins,
                const int d = t - s[g];          // matching sequential overwrite order
                if (d >= 0 && d < GAP_LEN) j = d;
            }
            if (j >= 0) {
                // coords = (j+0.5)*1.5 - 0.5 = 1.5j + 0.25, in [0.25, 23998.75]:
                // the reference clip is a no-op and trunc == floor.
                const float coord = 1.5f * (float)j + 0.25f;
                const int   i0    = (int)coord;
                const float w     = coord - (float)i0;
                const int   i1    = min(i0 + 1, GEN_N - 1);
                float val = grow[i0] * (1.0f - w) + grow[i1] * w;
                // Crossfade: linspace(0,1,1000)[j] = j/999 ; linspace(1,0,1000)[i] = 1 - i/999
                if (j < CF)
                    val *= (float)j * (1.0f / (float)(CF - 1));
                if (j >= GAP_LEN - CF)
                    val *= 1.0f - (float)(j - (GAP_LEN - CF)) * (1.0f / (float)(CF - 1));
                vals[e] = val;
            }
        }
        r.x = vals[0]; r.y = vals[1]; r.z = vals[2]; r.w = vals[3];
    }

    // 128-bit store; every element of d_out is written exactly once.
    *reinterpret_cast<float4*>(out + rowOff + t0) = r;
}

extern "C" void kernel_launch(void* const* d_in, const int* in_sizes, int n_in,
                              void* d_out, int out_size, void* d_ws, size_t ws_size,
                              hipStream_t stream)
{
    // setup_inputs() order: original_audio[f32 B*T], generated_audio[f32 B*GEN_N],
    // gap_starts[i32 B*N_GAPS], gap_length[i32 scalar, ==16000 by construction].
    const float* orig       = (const float*)d_in[0];
    const float* gen        = (const float*)d_in[1];
    const int*   gap_starts = (const int*)  d_in[2];
    float*       out        = (float*)d_out;

    (void)in_sizes; (void)n_in; (void)d_ws; (void)ws_size; (void)out_size;

    dim3 block(256);
    dim3 grid((T_LEN + 4 * 256 - 1) / (4 * 256), B_SZ);  // 1875 x 32 blocks
    audio_gap_fill_kernel<<<grid, block, 0, stream>>>(orig, gen, gap_starts, out);
}